// MultiHeadAttention_62697932587327
// MI455X (gfx1250) — compile-verified
//
#include <hip/hip_runtime.h>
#include <math.h>

// Fused cross-head entmax attention for MI455X (gfx1250, wave32).
// One wave processes 2 tokens (16 rows = 2 tokens x 8 heads) entirely through
// V_WMMA_F32_16X16X4_F32 stages: QKV projection, scores, attn@V.

typedef __attribute__((ext_vector_type(2))) float v2f;
typedef __attribute__((ext_vector_type(4))) float v4f;
typedef __attribute__((ext_vector_type(8))) float v8f;

#define WSTR 68   // padded row stride (floats) for 64-wide LDS tiles (bank-conflict free b64)
#define TSTR 68   // padded row stride for q/k/v 16x64 tiles
#define ASTR 20   // padded row stride for 16x16 score/attn tiles
#define WAVES_PER_BLOCK 4
#define THREADS (WAVES_PER_BLOCK * 32)

__device__ __forceinline__ v8f wmma_f32(v2f a, v2f b, v8f c) {
  // D(16x16,f32) = A(16x4,f32) * B(4x16,f32) + C
  return __builtin_amdgcn_wmma_f32_16x16x4_f32(
      /*neg_a=*/false, a, /*neg_b=*/false, b,
      /*c_mod=*/(short)0, c, /*reuse_a=*/false, /*reuse_b=*/false);
}

// One 16x64 = (16x64) @ (64x64)^T projection + bias, result to an LDS tile.
// ax[k]  : A fragment for K-chunk k (lane L holds rows L&15, K = 4k + {0,1}/{2,3})
// Wl     : LDS weight, row-major [e][d], row stride WSTR  (B[d][e] = W[e][d])
__device__ __forceinline__ void project_tile16x64(const v2f* ax, const float* Wl,
                                                  const float* bl, float* dst,
                                                  int l15, int half) {
#pragma unroll
  for (int t = 0; t < 4; ++t) {
    v8f acc = {0.f, 0.f, 0.f, 0.f, 0.f, 0.f, 0.f, 0.f};
    const float* wb = Wl + (t * 16 + l15) * WSTR + half * 2;
#pragma unroll
    for (int kk = 0; kk < 16; ++kk) {
      v2f bf = *(const v2f*)(wb + kk * 4);  // ds_load_b64, conflict-free (WSTR=68)
      acc = wmma_f32(ax[kk], bf, acc);
    }
    float bias = bl[t * 16 + l15];
#pragma unroll
    for (int i = 0; i < 8; ++i)
      dst[(i + half * 8) * TSTR + t * 16 + l15] = acc[i] + bias;
  }
}

#define CSWAP(i, j)                          \
  {                                          \
    float lo_ = fminf(xs[i], xs[j]);         \
    float hi_ = fmaxf(xs[i], xs[j]);         \
    xs[i] = hi_;                             \
    xs[j] = lo_;                             \
  }

__global__ __launch_bounds__(THREADS) void fused_crosshead_attn(
    const float* __restrict__ x, const float* __restrict__ Wq,
    const float* __restrict__ bq, const float* __restrict__ Wk,
    const float* __restrict__ bk, const float* __restrict__ Wv,
    const float* __restrict__ bv, float* __restrict__ out, int npairs) {
  __shared__ float sW[3][64 * WSTR];                 // padded Wq/Wk/Wv
  __shared__ float sBias[3][64];
  __shared__ float sQV[WAVES_PER_BLOCK][16 * TSTR];  // q tile, later reused for v
  __shared__ float sK[WAVES_PER_BLOCK][16 * TSTR];
  __shared__ float sS[WAVES_PER_BLOCK][16 * ASTR];   // scores
  __shared__ float sA[WAVES_PER_BLOCK][16 * ASTR];   // block-diagonal attn

  const int tid = threadIdx.x;

  // Cooperative, padded weight load (once per block).
  {
    const float* Ws[3] = {Wq, Wk, Wv};
    const float* Bs[3] = {bq, bk, bv};
#pragma unroll
    for (int p = 0; p < 3; ++p) {
      for (int i = tid; i < 64 * 64; i += THREADS) {
        int e = i >> 6, d = i & 63;
        sW[p][e * WSTR + d] = Ws[p][i];
      }
      if (tid < 64) sBias[p][tid] = Bs[p][tid];
    }
  }
  __syncthreads();

  const int wave   = (int)blockIdx.x * WAVES_PER_BLOCK + (tid >> 5);
  const int stride = (int)gridDim.x * WAVES_PER_BLOCK;
  const int lane = tid & 31;
  const int half = lane >> 4;   // 0: lanes 0-15, 1: lanes 16-31
  const int l15  = lane & 15;
  const int w    = tid >> 5;

  float* q_lds = sQV[w];
  float* k_lds = sK[w];
  float* s_lds = sS[w];
  float* a_lds = sA[w];

  const float SCALE = 0.044194173824159216f;  // 1/sqrt(EMBED_SIZE=512)

  for (int pair = wave; pair < npairs; pair += stride) {
    const float* xp = x + (size_t)pair * 1024;  // 2 tokens * 512 floats
    if (pair + stride < npairs)                 // prefetch next pair's x (4KB)
      __builtin_prefetch(x + (size_t)(pair + stride) * 1024 + lane * 32, 0, 1);

    // ---- A fragments of X: 16 rows (2 tok x 8 heads) x K=64, b64 per chunk.
    v2f ax[16];
    const float* xr = xp + l15 * 64 + half * 2;
#pragma unroll
    for (int kk = 0; kk < 16; ++kk) ax[kk] = *(const v2f*)(xr + kk * 4);

    // ---- Q and K projections (WMMA), tiles to LDS.
    project_tile16x64(ax, sW[0], sBias[0], q_lds, l15, half);
    project_tile16x64(ax, sW[1], sBias[1], k_lds, l15, half);
    __builtin_amdgcn_wave_barrier();

    // ---- scores = (q @ k^T) * scale : 16x16, K=64.  B[d][g] = k[g][d].
    {
      v8f acc = {0.f, 0.f, 0.f, 0.f, 0.f, 0.f, 0.f, 0.f};
      const float* qa = q_lds + l15 * TSTR + half * 2;
      const float* kb = k_lds + l15 * TSTR + half * 2;
#pragma unroll
      for (int kk = 0; kk < 16; ++kk) {
        v2f a = *(const v2f*)(qa + kk * 4);
        v2f b = *(const v2f*)(kb + kk * 4);
        acc = wmma_f32(a, b, acc);
      }
#pragma unroll
      for (int i = 0; i < 8; ++i)
        s_lds[(i + half * 8) * ASTR + l15] = acc[i] * SCALE;
    }
    __builtin_amdgcn_wave_barrier();

    // ---- exact 1.5-entmax over the 8 heads of each of the 16 rows.
    if (lane < 16) {
      const int m = lane;
      const int g0 = (m < 8) ? 0 : 8;  // valid 8x8 diagonal block for this token
      const float* sr = s_lds + m * ASTR + g0;
      float xv[8], xs[8];
      float mx = sr[0];
#pragma unroll
      for (int j = 1; j < 8; ++j) mx = fmaxf(mx, sr[j]);
#pragma unroll
      for (int j = 0; j < 8; ++j) {
        xv[j] = (sr[j] - mx) * 0.5f;
        xs[j] = xv[j];
      }
      // descending sort, 19-comparator Batcher network
      CSWAP(0, 1) CSWAP(2, 3) CSWAP(4, 5) CSWAP(6, 7)
      CSWAP(0, 2) CSWAP(1, 3) CSWAP(4, 6) CSWAP(5, 7)
      CSWAP(1, 2) CSWAP(5, 6) CSWAP(0, 4) CSWAP(3, 7)
      CSWAP(1, 5) CSWAP(2, 6)
      CSWAP(1, 4) CSWAP(3, 6)
      CSWAP(2, 4) CSWAP(3, 5)
      CSWAP(3, 4)
      float cum = 0.f, cumsq = 0.f, tau_star = 0.f;
#pragma unroll
      for (int k2 = 1; k2 <= 8; ++k2) {
        float vk = xs[k2 - 1];
        cum += vk;
        cumsq += vk * vk;
        float rk = 1.0f / (float)k2;
        float mean = cum * rk;
        float ss = cumsq - cum * mean;     // = k*(meansq - mean^2)
        float delta = (1.0f - ss) * rk;
        float sq = delta > 0.f ? __builtin_sqrtf(delta) : 0.f;
        float tau = mean - sq;
        if (tau <= vk) tau_star = tau;     // last k with tau_k <= xs_k == support
      }
      // write block-diagonal attn row (off-block zeroed for the out-WMMA)
      float* ar = a_lds + m * ASTR;
      v4f z = {0.f, 0.f, 0.f, 0.f};
      *(v4f*)(ar + (8 - g0)) = z;
      *(v4f*)(ar + (8 - g0) + 4) = z;
#pragma unroll
      for (int j = 0; j < 8; ++j) {
        float t0 = fmaxf(xv[j] - tau_star, 0.f);
        ar[g0 + j] = t0 * t0;
      }
    }
    __builtin_amdgcn_wave_barrier();

    // ---- V projection (reuses q tile buffer; q already consumed).
    project_tile16x64(ax, sW[2], sBias[2], q_lds, l15, half);
    __builtin_amdgcn_wave_barrier();

    // ---- out = attn(16x16 block-diag) @ v(16x64), K=16 (4 WMMA chunks).
    float* op = out + (size_t)pair * 1024;
    const float* aa = a_lds + l15 * ASTR + half * 2;
#pragma unroll
    for (int t = 0; t < 4; ++t) {
      v8f acc = {0.f, 0.f, 0.f, 0.f, 0.f, 0.f, 0.f, 0.f};
#pragma unroll
      for (int kk = 0; kk < 4; ++kk) {
        v2f a = *(const v2f*)(aa + kk * 4);
        int r0 = kk * 4 + half * 2;  // K rows of v tile (row-major [g][d])
        v2f b;
        b.x = q_lds[(r0)*TSTR + t * 16 + l15];
        b.y = q_lds[(r0 + 1) * TSTR + t * 16 + l15];
        acc = wmma_f32(a, b, acc);
      }
#pragma unroll
      for (int i = 0; i < 8; ++i)
        op[(size_t)(i + half * 8) * 64 + t * 16 + l15] = acc[i];
    }
  }
}

extern "C" void kernel_launch(void* const* d_in, const int* in_sizes, int n_in,
                              void* d_out, int out_size, void* d_ws, size_t ws_size,
                              hipStream_t stream) {
  (void)n_in; (void)out_size; (void)d_ws; (void)ws_size;
  const float* x  = (const float*)d_in[0];
  const float* Wq = (const float*)d_in[1];
  const float* bq = (const float*)d_in[2];
  const float* Wk = (const float*)d_in[3];
  const float* bk = (const float*)d_in[4];
  const float* Wv = (const float*)d_in[5];
  const float* bv = (const float*)d_in[6];
  float* out = (float*)d_out;

  const int ntokens = in_sizes[0] / 512;  // C*T
  const int npairs = ntokens >> 1;        // 2 tokens per wave iteration

  dim3 grid(1024), block(THREADS);
  hipLaunchKernelGGL(fused_crosshead_attn, grid, block, 0, stream,
                     x, Wq, bq, Wk, bk, Wv, bv, out, npairs);
}